// SuperPointMatchesGenerator_9019431321862
// MI455X (gfx1250) — compile-verified
//
#include <hip/hip_runtime.h>
#include <math.h>

// Problem constants (match reference)
#define B_ 4
#define N_ 4096
#define M_ 4096
static_assert(N_ == M_, "prep kernel assumes N == M");

#define IMG_W_ 1024.0f
#define IMG_H_ 1024.0f
#define POS_T_ 3.0f
#define NEG_T_ 5.0f

typedef float v2f __attribute__((ext_vector_type(2)));
typedef float v8f __attribute__((ext_vector_type(8)));

// ---------------------------------------------------------------------------
// 3x3 inverse (adjugate / det)
// ---------------------------------------------------------------------------
__device__ __forceinline__ void invert3x3(const float* H, float* Hi) {
    float a = H[0], b = H[1], c = H[2];
    float d = H[3], e = H[4], f = H[5];
    float g = H[6], h = H[7], i = H[8];
    float A  =  (e * i - f * h);
    float Bc = -(d * i - f * g);
    float Cc =  (d * h - e * g);
    float inv = 1.0f / (a * A + b * Bc + c * Cc);
    Hi[0] = A * inv;
    Hi[1] = -(b * i - c * h) * inv;
    Hi[2] =  (b * f - c * e) * inv;
    Hi[3] = Bc * inv;
    Hi[4] =  (a * i - c * g) * inv;
    Hi[5] = -(a * f - c * d) * inv;
    Hi[6] = Cc * inv;
    Hi[7] = -(a * h - b * g) * inv;
    Hi[8] =  (a * e - b * d) * inv;
}

__device__ __forceinline__ void reproject(const float* Hm, float x, float y,
                                          float& tx, float& ty, int& mask) {
    float px = Hm[0] * x + Hm[1] * y + Hm[2];
    float py = Hm[3] * x + Hm[4] * y + Hm[5];
    float pz = Hm[6] * x + Hm[7] * y + Hm[8];
    float zc = (fabsf(pz) < 1e-8f) ? 1e-8f : pz;
    tx = px / zc;
    ty = py / zc;
    bool inb = (tx >= 0.0f) && (tx < IMG_W_) && (ty >= 0.0f) && (ty < IMG_H_);
    mask = ((pz > 1e-3f) && inb) ? 1 : 0;
}

// ---------------------------------------------------------------------------
// Kernel 1: reprojection + WMMA operand packing.
//   A-side row  (transformed pt):  ( ax,  ay, ax^2+ay^2, 1 )
//   B-side col  (raw pt)        :  (-2bx, -2by, 1, bx^2+by^2)
// A-row . B-col == full squared distance (K = 4 of V_WMMA_F32_16X16X4_F32).
// ---------------------------------------------------------------------------
__global__ __launch_bounds__(256) void prep_kernel(
    const float* __restrict__ kpts0, const float* __restrict__ kpts1,
    const float* __restrict__ Hbuf,
    float* __restrict__ A0,  float* __restrict__ Bm0,   // d01: A from kpts0_t, B from kpts1
    float* __restrict__ A1,  float* __restrict__ Bm1,   // d10: A from kpts1_t, B from kpts0
    int* __restrict__ mask0) {
    int gidx = blockIdx.x * blockDim.x + threadIdx.x;
    if (gidx >= B_ * N_) return;
    int b = gidx / N_;
    int i = gidx - b * N_;

    float H[9], Hi[9];
#pragma unroll
    for (int k = 0; k < 9; ++k) H[k] = Hbuf[b * 9 + k];
    invert3x3(H, Hi);

    // ---- kpts0: transform by H -> A0 ; raw -> Bm1 ----
    float x0 = kpts0[(size_t)(b * N_ + i) * 2 + 0];
    float y0 = kpts0[(size_t)(b * N_ + i) * 2 + 1];
    float tx0, ty0; int m0;
    reproject(H, x0, y0, tx0, ty0, m0);
    mask0[b * N_ + i] = m0;
    float4 a0 = make_float4(tx0, ty0, tx0 * tx0 + ty0 * ty0, 1.0f);
    *(float4*)(A0 + (size_t)(b * N_ + i) * 4) = a0;
    float4 bm1 = make_float4(-2.0f * x0, -2.0f * y0, 1.0f, x0 * x0 + y0 * y0);
    *(float4*)(Bm1 + (size_t)(b * N_ + i) * 4) = bm1;

    // ---- kpts1: transform by H^-1 -> A1 ; raw -> Bm0 ---- (mask1 unused by ref)
    float x1 = kpts1[(size_t)(b * M_ + i) * 2 + 0];
    float y1 = kpts1[(size_t)(b * M_ + i) * 2 + 1];
    float tx1, ty1; int m1;
    reproject(Hi, x1, y1, tx1, ty1, m1);
    (void)m1;
    float4 a1 = make_float4(tx1, ty1, tx1 * tx1 + ty1 * ty1, 1.0f);
    *(float4*)(A1 + (size_t)(b * M_ + i) * 4) = a1;
    float4 bm0 = make_float4(-2.0f * x1, -2.0f * y1, 1.0f, x1 * x1 + y1 * y1);
    *(float4*)(Bm0 + (size_t)(b * M_ + i) * 4) = bm0;
}

// ---------------------------------------------------------------------------
// Kernel 2: WMMA-based row-wise min/argmin of the RxC squared-distance matrix.
// One wave32 owns a 16-row tile and sweeps C in 16-column chunks; each chunk
// is one v_wmma_f32_16x16x4_f32 producing 256 squared distances.
// D layout (ISA 7.12.2): VGPR r -> row r (lanes 0-15) / row r+8 (lanes 16-31),
// N = lane & 15.  Argmin finished with a 16-lane shuffle reduction.
// ---------------------------------------------------------------------------
__global__ __launch_bounds__(256) void argmin_kernel(
    const float* __restrict__ Adata,  // [B][R][4] packed A rows
    const float* __restrict__ Bdata,  // [B][C][4] packed B cols
    int* __restrict__ gm, float* __restrict__ mdsq,
    int R, int C) {
    int lane = threadIdx.x & 31;
    int wave = threadIdx.x >> 5;
    int wavesPerBlock = blockDim.x >> 5;
    int tile = blockIdx.x * wavesPerBlock + wave;
    int tilesPerBatch = R >> 4;
    int b = tile / tilesPerBatch;
    if (b >= B_) return;
    int rowBase = (tile - b * tilesPerBatch) << 4;

    const float* Ab = Adata + (size_t)b * R * 4;
    const float* Bb = Bdata + (size_t)b * C * 4;

    int  l16 = lane & 15;
    bool hi  = lane >= 16;

    // A operand (16x4 f32, 2 VGPRs): lanes 0-15 hold K=0,1; lanes 16-31 K=2,3.
    float4 av = *(const float4*)(Ab + (size_t)(rowBase + l16) * 4);
    v2f a;
    a.x = hi ? av.z : av.x;
    a.y = hi ? av.w : av.y;

    float minv[8];
    int   mini[8];
#pragma unroll
    for (int r = 0; r < 8; ++r) { minv[r] = 3.4028235e38f; mini[r] = 0; }

    for (int c0 = 0; c0 < C; c0 += 16) {
        float4 bv = *(const float4*)(Bb + (size_t)(c0 + l16) * 4);
        v2f bb;
        bb.x = hi ? bv.z : bv.x;
        bb.y = hi ? bv.w : bv.y;

        v8f d;
#if __has_builtin(__builtin_amdgcn_wmma_f32_16x16x4_f32)
        v8f cz = {};
        d = __builtin_amdgcn_wmma_f32_16x16x4_f32(
            /*neg_a=*/false, a, /*neg_b=*/false, bb,
            /*c_mod=*/(short)0, cz, /*reuse_a=*/false, /*reuse_b=*/false);
#else
        // VALU fallback: every lane holds full packed row/col; broadcast A rows.
#pragma unroll
        for (int r = 0; r < 8; ++r) {
            int src = r + (hi ? 8 : 0);
            float arx = __shfl(av.x, src, 32);
            float ary = __shfl(av.y, src, 32);
            float arz = __shfl(av.z, src, 32);
            float arw = __shfl(av.w, src, 32);
            d[r] = arx * bv.x + ary * bv.y + arz * bv.z + arw * bv.w;
        }
#endif
        int col = c0 + l16;
#pragma unroll
        for (int r = 0; r < 8; ++r) {
            float dv = (float)d[r];
            bool lt = dv < minv[r];       // strict < keeps first index in-lane
            minv[r] = lt ? dv : minv[r];
            mini[r] = lt ? col : mini[r];
        }
    }

    // Cross-lane reduction within each 16-lane half (rows r and r+8).
#pragma unroll
    for (int r = 0; r < 8; ++r) {
        float mv = minv[r];
        int   mi = mini[r];
#pragma unroll
        for (int off = 8; off >= 1; off >>= 1) {
            float ov = __shfl_xor(mv, off, 16);
            int   oi = __shfl_xor(mi, off, 16);
            if (ov < mv || (ov == mv && oi < mi)) { mv = ov; mi = oi; }
        }
        if (l16 == 0) {
            int row = rowBase + r + (hi ? 8 : 0);
            gm[(size_t)b * R + row]   = mi;
            mdsq[(size_t)b * R + row] = mv;
        }
    }
}

// ---------------------------------------------------------------------------
// Kernel 3: base gt1 / min_dist1  (must run BEFORE finalize0's scatter)
// ---------------------------------------------------------------------------
__global__ __launch_bounds__(256) void finalize1_kernel(
    const float* __restrict__ md1sq,
    float* __restrict__ out_gt1, float* __restrict__ out_md1) {
    int g = blockIdx.x * blockDim.x + threadIdx.x;
    if (g >= B_ * M_) return;
    float md = sqrtf(fmaxf(md1sq[g], 1e-12f));
    out_md1[g] = md;
    out_gt1[g] = (md > NEG_T_) ? -1.0f : -2.0f;
}

// ---------------------------------------------------------------------------
// Kernel 4: gt0 thresholds + mutual-consistency + injective scatter into gt1
// ---------------------------------------------------------------------------
__global__ __launch_bounds__(256) void finalize0_kernel(
    const int* __restrict__ gm0, const int* __restrict__ gm1,
    const float* __restrict__ md0sq, const int* __restrict__ mask0,
    float* __restrict__ out_gt0, float* __restrict__ out_gt1,
    float* __restrict__ out_md0) {
    int g = blockIdx.x * blockDim.x + threadIdx.x;
    if (g >= B_ * N_) return;
    int b = g / N_;
    int i = g - b * N_;

    int m    = gm0[g];
    int back = gm1[(size_t)b * M_ + m];
    float md = sqrtf(fmaxf(md0sq[g], 1e-12f));

    int gt = (back != i) ? -1 : m;
    if (!mask0[g])  gt = -2;
    if (md > POS_T_) gt = -2;
    if (md > NEG_T_) gt = -1;

    out_md0[g] = md;
    out_gt0[g] = (float)gt;
    if (gt >= 0)  // mutual consistency => injective mapping, plain store is safe
        out_gt1[(size_t)b * M_ + gt] = (float)i;
}

// ---------------------------------------------------------------------------
extern "C" void kernel_launch(void* const* d_in, const int* in_sizes, int n_in,
                              void* d_out, int out_size, void* d_ws, size_t ws_size,
                              hipStream_t stream) {
    const float* kpts0 = (const float*)d_in[0];   // [B,N,2]
    const float* kpts1 = (const float*)d_in[1];   // [B,M,2]
    const float* Hbuf  = (const float*)d_in[2];   // [B,3,3]

    float* out = (float*)d_out;
    // Output order: gt0 [B*N], gt1 [B*M], min_dist0 [B*N], min_dist1 [B*M]
    float* out_gt0 = out;
    float* out_gt1 = out + B_ * N_;
    float* out_md0 = out + B_ * N_ + B_ * M_;
    float* out_md1 = out + 2 * B_ * N_ + B_ * M_;

    // Workspace layout (4-byte units)
    float* ws    = (float*)d_ws;
    float* A0    = ws;                       // B*N*4
    float* Bm0   = A0  + (size_t)B_ * N_ * 4; // B*M*4
    float* A1    = Bm0 + (size_t)B_ * M_ * 4; // B*M*4
    float* Bm1   = A1  + (size_t)B_ * M_ * 4; // B*N*4
    float* md0sq = Bm1 + (size_t)B_ * N_ * 4; // B*N
    float* md1sq = md0sq + (size_t)B_ * N_;   // B*M
    int*   mask0 = (int*)(md1sq + (size_t)B_ * M_); // B*N
    int*   gm0   = mask0 + (size_t)B_ * N_;         // B*N
    int*   gm1   = gm0   + (size_t)B_ * N_;         // B*M

    // 1) reproject + pack WMMA operands
    {
        int threads = 256;
        int blocks = (B_ * N_ + threads - 1) / threads;
        prep_kernel<<<blocks, threads, 0, stream>>>(kpts0, kpts1, Hbuf,
                                                    A0, Bm0, A1, Bm1, mask0);
    }
    // 2) d01 argmin: rows = kpts0_t, cols = kpts1
    {
        int wavesPerBlock = 8;                         // 256 threads
        int tiles = B_ * (N_ / 16);
        int blocks = (tiles + wavesPerBlock - 1) / wavesPerBlock;
        argmin_kernel<<<blocks, wavesPerBlock * 32, 0, stream>>>(
            A0, Bm0, gm0, md0sq, N_, M_);
    }
    // 3) d10 argmin: rows = kpts1_t, cols = kpts0
    {
        int wavesPerBlock = 8;
        int tiles = B_ * (M_ / 16);
        int blocks = (tiles + wavesPerBlock - 1) / wavesPerBlock;
        argmin_kernel<<<blocks, wavesPerBlock * 32, 0, stream>>>(
            A1, Bm1, gm1, md1sq, M_, N_);
    }
    // 4) base gt1 / min_dist1 (before scatter)
    {
        int threads = 256;
        int blocks = (B_ * M_ + threads - 1) / threads;
        finalize1_kernel<<<blocks, threads, 0, stream>>>(md1sq, out_gt1, out_md1);
    }
    // 5) gt0 + scatter into gt1
    {
        int threads = 256;
        int blocks = (B_ * N_ + threads - 1) / threads;
        finalize0_kernel<<<blocks, threads, 0, stream>>>(
            gm0, gm1, md0sq, mask0, out_gt0, out_gt1, out_md0);
    }
}